// HierarchicalBoxEmbeddingsPrior_68831145886680
// MI455X (gfx1250) — compile-verified
//
#include <hip/hip_runtime.h>
#include <hip/hip_bf16.h>
#include <math.h>

typedef float v2f __attribute__((ext_vector_type(2)));
typedef float v8f __attribute__((ext_vector_type(8)));

#define EPS10 1e-10f

// ---------------- Output layout (floats, return order) ----------------
// 0      : root_mu_min   (64x128)   8192
// 8192   : root_mu_max              8192
// 16384  : beta_min0                8192
// 24576  : beta_max0                8192
// 32768  : mu_min1       (512x128)  65536
// 98304  : mu_max1                  65536
// 163840 : b_min1                   65536
// 229376 : b_max1                   65536
// 294912 : mu_min2       (2048x128) 262144
// 557056 : mu_max2                  262144
// 819200 : b_min2                   262144
// 1081344: b_max2                   262144   (total 1343488)

// ---------------------------------------------------------------------
// Level-0: copy mu, beta = sigmoid(raw) + 1e-6
__global__ __launch_bounds__(256)
void level0_kernel(const float* __restrict__ mu_min, const float* __restrict__ mu_max,
                   const float* __restrict__ bmin_raw, const float* __restrict__ bmax_raw,
                   float* __restrict__ out)
{
    int i = blockIdx.x * blockDim.x + threadIdx.x;
    if (i >= 64 * 128) return;
    out[i]          = mu_min[i];
    out[8192  + i]  = mu_max[i];
    out[16384 + i]  = 1.0f / (1.0f + expf(-bmin_raw[i])) + 1e-6f;
    out[24576 + i]  = 1.0f / (1.0f + expf(-bmax_raw[i])) + 1e-6f;
}

// ---------------------------------------------------------------------
// Hard gumbel-sigmoid: w = (logits + g > 0), plus log(w+eps) and 1/(rowsum+eps).
// One block per row c.
__global__ void make_w_kernel(const float* __restrict__ logits, const float* __restrict__ u,
                              float* __restrict__ w, float* __restrict__ logw,
                              float* __restrict__ inv_wsum, int P)
{
    __shared__ float red[256];
    const int c = blockIdx.x;
    const int tid = threadIdx.x;
    float part = 0.0f;
    for (int p = tid; p < P; p += blockDim.x) {
        const size_t idx = (size_t)c * P + p;
        float g  = -logf(-logf(u[idx] + EPS10) + EPS10);
        float wv = (logits[idx] + g) > 0.0f ? 1.0f : 0.0f;
        w[idx]    = wv;
        logw[idx] = logf(wv + EPS10);   // 0 or log(1e-10)
        part += wv;
    }
    red[tid] = part;
    __syncthreads();
    for (int s = blockDim.x >> 1; s > 0; s >>= 1) {
        if (tid < s) red[tid] += red[tid + s];
        __syncthreads();
    }
    if (tid == 0) inv_wsum[c] = 1.0f / (red[0] + EPS10);
}

// ---------------------------------------------------------------------
// b_min[c,d] = (sum_p w[c,p]*beta_min[p,d]) * inv_wsum[c]   (same for max)
// One wave computes a 32x32 (c,d) output tile for BOTH matrices:
//   2 C-sub-tiles x 2 D-sub-tiles x {min,max} = 8 accumulators (64 VGPRs),
//   per K=4 step: 2 A-loads + 4 B-loads -> 8 x V_WMMA_F32_16X16X4_F32.
// A (16x4 f32): lane<16 -> M=lane, K={0,1}; lane>=16 -> M=lane-16, K={2,3}
// B (4x16 f32): mirrored; C/D: VGPR r -> rows {r, r+8} split across halves.
// P is compile-time so the K-loop has no runtime guards.
template <int P>
__global__ __launch_bounds__(32)
void wmma_bmat_kernel(const float* __restrict__ w,        // C x P
                      const float* __restrict__ beta_min, // P x D
                      const float* __restrict__ beta_max, // P x D
                      const float* __restrict__ inv_wsum, // C
                      float* __restrict__ b_min,          // C x D
                      float* __restrict__ b_max,          // C x D
                      int D)
{
    const int td = blockIdx.x << 5;      // 32-wide D tile
    const int tc = blockIdx.y << 5;      // 32-tall C tile
    const int lane = threadIdx.x;
    const int half = lane >> 4;          // 0 | 1
    const int l16  = lane & 15;

    v8f accMin0t0 = {}, accMin0t1 = {}, accMin1t0 = {}, accMin1t1 = {};
    v8f accMax0t0 = {}, accMax0t1 = {}, accMax1t0 = {}, accMax1t1 = {};

    const size_t arow0 = (size_t)(tc + l16) * P;
    const size_t arow1 = (size_t)(tc + 16 + l16) * P;
    const int col0 = td + l16;
    const int col1 = td + 16 + l16;

    #pragma unroll 4
    for (int p0 = 0; p0 < P; p0 += 4) {
        const int k0 = p0 + half * 2;
        // A tiles (8-byte aligned pair loads)
        v2f a0 = *(const v2f*)(w + arow0 + k0);
        v2f a1 = *(const v2f*)(w + arow1 + k0);
        // B tiles (two D sub-tiles, min & max)
        v2f bn0, bn1, bx0, bx1;
        bn0.x = beta_min[(size_t)(k0 + 0) * D + col0];
        bn0.y = beta_min[(size_t)(k0 + 1) * D + col0];
        bn1.x = beta_min[(size_t)(k0 + 0) * D + col1];
        bn1.y = beta_min[(size_t)(k0 + 1) * D + col1];
        bx0.x = beta_max[(size_t)(k0 + 0) * D + col0];
        bx0.y = beta_max[(size_t)(k0 + 1) * D + col0];
        bx1.x = beta_max[(size_t)(k0 + 0) * D + col1];
        bx1.y = beta_max[(size_t)(k0 + 1) * D + col1];

        accMin0t0 = __builtin_amdgcn_wmma_f32_16x16x4_f32(false, a0, false, bn0, (short)0, accMin0t0, false, false);
        accMin0t1 = __builtin_amdgcn_wmma_f32_16x16x4_f32(false, a0, false, bn1, (short)0, accMin0t1, false, false);
        accMin1t0 = __builtin_amdgcn_wmma_f32_16x16x4_f32(false, a1, false, bn0, (short)0, accMin1t0, false, false);
        accMin1t1 = __builtin_amdgcn_wmma_f32_16x16x4_f32(false, a1, false, bn1, (short)0, accMin1t1, false, false);
        accMax0t0 = __builtin_amdgcn_wmma_f32_16x16x4_f32(false, a0, false, bx0, (short)0, accMax0t0, false, false);
        accMax0t1 = __builtin_amdgcn_wmma_f32_16x16x4_f32(false, a0, false, bx1, (short)0, accMax0t1, false, false);
        accMax1t0 = __builtin_amdgcn_wmma_f32_16x16x4_f32(false, a1, false, bx0, (short)0, accMax1t0, false, false);
        accMax1t1 = __builtin_amdgcn_wmma_f32_16x16x4_f32(false, a1, false, bx1, (short)0, accMax1t1, false, false);
    }

    #pragma unroll
    for (int r = 0; r < 8; ++r) {
        const int row0 = tc + r + half * 8;
        const int row1 = tc + 16 + r + half * 8;
        const float s0 = inv_wsum[row0];
        const float s1 = inv_wsum[row1];
        b_min[(size_t)row0 * D + col0] = accMin0t0[r] * s0;
        b_min[(size_t)row0 * D + col1] = accMin0t1[r] * s0;
        b_min[(size_t)row1 * D + col0] = accMin1t0[r] * s1;
        b_min[(size_t)row1 * D + col1] = accMin1t1[r] * s1;
        b_max[(size_t)row0 * D + col0] = accMax0t0[r] * s0;
        b_max[(size_t)row0 * D + col1] = accMax0t1[r] * s0;
        b_max[(size_t)row1 * D + col0] = accMax1t0[r] * s1;
        b_max[(size_t)row1 * D + col1] = accMax1t1[r] * s1;
    }
}

// ---------------------------------------------------------------------
// mu_min_o[c,d] =  b_min[c,d] * LSE_p( logw[c,p] + mu_min_p[p,d]/b_min[c,d] )
// mu_max_o[c,d] = -b_max[c,d] * LSE_p( logw[c,p] - mu_max_p[p,d]/b_max[c,d] )
// Block per c (128 threads = one d each); logw row staged in LDS; two-pass
// max-stabilized LSE, mu reads coalesced (stride-D rows), L2 resident.
__global__ __launch_bounds__(128)
void lse_kernel(const float* __restrict__ logw,      // C x P
                const float* __restrict__ mu_min_p,  // P x D
                const float* __restrict__ mu_max_p,  // P x D
                const float* __restrict__ b_min,     // C x D
                const float* __restrict__ b_max,     // C x D
                float* __restrict__ mu_min_o,        // C x D
                float* __restrict__ mu_max_o,        // C x D
                int P, int D)
{
    extern __shared__ float slogw[];
    const int c = blockIdx.x;
    const int d = threadIdx.x;
    for (int p = threadIdx.x; p < P; p += blockDim.x)
        slogw[p] = logw[(size_t)c * P + p];
    __syncthreads();

    const float bmn = b_min[(size_t)c * D + d];
    const float bmx = b_max[(size_t)c * D + d];
    const float inv_bmn = 1.0f / bmn;
    const float inv_bmx = 1.0f / bmx;

    float m1 = -3.402823466e38f, m2 = -3.402823466e38f;
    for (int p = 0; p < P; ++p) {
        const float lw = slogw[p];
        m1 = fmaxf(m1, fmaf(mu_min_p[(size_t)p * D + d],  inv_bmn, lw));
        m2 = fmaxf(m2, fmaf(mu_max_p[(size_t)p * D + d], -inv_bmx, lw));
    }
    float s1 = 0.0f, s2 = 0.0f;
    for (int p = 0; p < P; ++p) {
        const float lw = slogw[p];
        s1 += expf(fmaf(mu_min_p[(size_t)p * D + d],  inv_bmn, lw) - m1);
        s2 += expf(fmaf(mu_max_p[(size_t)p * D + d], -inv_bmx, lw) - m2);
    }
    mu_min_o[(size_t)c * D + d] =  bmn * (logf(s1) + m1);
    mu_max_o[(size_t)c * D + d] = -bmx * (logf(s2) + m2);
}

// ---------------------------------------------------------------------
extern "C" void kernel_launch(void* const* d_in, const int* in_sizes, int n_in,
                              void* d_out, int out_size, void* d_ws, size_t ws_size,
                              hipStream_t stream)
{
    const float* root_mu_min   = (const float*)d_in[0];
    const float* root_mu_max   = (const float*)d_in[1];
    const float* root_bmin_raw = (const float*)d_in[2];
    const float* root_bmax_raw = (const float*)d_in[3];
    const float* adj0          = (const float*)d_in[4];   // 512 x 64
    const float* adj1          = (const float*)d_in[5];   // 2048 x 512
    const float* u0            = (const float*)d_in[6];
    const float* u1            = (const float*)d_in[7];
    float* out = (float*)d_out;
    float* ws  = (float*)d_ws;

    // workspace layout (floats)
    float* w0     = ws;                       // 512*64
    float* logw0  = ws + 32768;               // 512*64
    float* iws0   = ws + 65536;               // 512
    float* w1     = ws + 66048;               // 2048*512
    float* logw1  = ws + 66048 + 1048576;     // 2048*512
    float* iws1   = ws + 66048 + 2097152;     // 2048

    float* beta_min0 = out + 16384;
    float* beta_max0 = out + 24576;
    float* mu_min1   = out + 32768;
    float* mu_max1   = out + 98304;
    float* b_min1    = out + 163840;
    float* b_max1    = out + 229376;
    float* mu_min2   = out + 294912;
    float* mu_max2   = out + 557056;
    float* b_min2    = out + 819200;
    float* b_max2    = out + 1081344;

    // level 0
    level0_kernel<<<(8192 + 255) / 256, 256, 0, stream>>>(
        root_mu_min, root_mu_max, root_bmin_raw, root_bmax_raw, out);

    // level 1 : C=512, P=64, D=128
    make_w_kernel<<<512, 64, 0, stream>>>(adj0, u0, w0, logw0, iws0, 64);
    wmma_bmat_kernel<64><<<dim3(128 / 32, 512 / 32), 32, 0, stream>>>(
        w0, beta_min0, beta_max0, iws0, b_min1, b_max1, 128);
    lse_kernel<<<512, 128, 64 * sizeof(float), stream>>>(
        logw0, root_mu_min, root_mu_max, b_min1, b_max1, mu_min1, mu_max1, 64, 128);

    // level 2 : C=2048, P=512, D=128
    make_w_kernel<<<2048, 256, 0, stream>>>(adj1, u1, w1, logw1, iws1, 512);
    wmma_bmat_kernel<512><<<dim3(128 / 32, 2048 / 32), 32, 0, stream>>>(
        w1, b_min1, b_max1, iws1, b_min2, b_max2, 128);
    lse_kernel<<<2048, 128, 512 * sizeof(float), stream>>>(
        logw1, mu_min1, mu_max1, b_min2, b_max2, mu_min2, mu_max2, 512, 128);
}